// KNNMemory_29111288332316
// MI455X (gfx1250) — compile-verified
//
#include <hip/hip_runtime.h>

// Problem constants (fixed by reference setup_inputs)
#define BATCH  16
#define NQ     512
#define MKEYS  16384
#define DIM    64
#define TOPK   32
#define CHUNK  64            // keys per LDS chunk (double buffered)
#define NCHUNK (MKEYS / CHUNK)
#define QTILE  16            // queries per workgroup
#define PADD   (DIM + 4)     // padded LDS row (floats) -> conflict-free strided reads
#define SPAD   (CHUNK + 4)   // padded score row

typedef __attribute__((ext_vector_type(2))) float v2f;
typedef __attribute__((ext_vector_type(4))) float v4f;
typedef __attribute__((ext_vector_type(8))) float v8f;

// CDNA5 async global->LDS copy (ASYNCcnt-tracked), per-lane 16 bytes.
__device__ __forceinline__ void async_copy_b128(unsigned lds_off, const float* gsrc) {
    unsigned long long ga = (unsigned long long)(uintptr_t)gsrc;
    asm volatile("global_load_async_to_lds_b128 %0, %1, off"
                 :: "v"(lds_off), "v"(ga)
                 : "memory");
}
__device__ __forceinline__ void wait_async0() {
    asm volatile("s_wait_asynccnt 0" ::: "memory");
}

__global__ __launch_bounds__(128) void knn_topk_kernel(
    const float* __restrict__ q, const float* __restrict__ keys,
    const float* __restrict__ db, float* __restrict__ out_kv,
    float* __restrict__ out_mask)
{
    __shared__ float qs[QTILE][PADD];        // query tile
    __shared__ float kc[2][CHUNK][PADD];     // double-buffered key chunk
    __shared__ float sc[QTILE][SPAD];        // score tile 16 x 64
    __shared__ int   top_idx[QTILE][TOPK];

    const int tid  = threadIdx.x;
    const int lane = tid & 31;
    const int wave = tid >> 5;
    const int b    = blockIdx.x >> 5;        // 32 query tiles per batch
    const int qt   = blockIdx.x & 31;
    const int q0   = qt * QTILE;

    const float* kbatch = keys + (size_t)b * MKEYS * DIM;

    // ---- prologue: kick off async copy of chunk 0 into buffer 0 ----
    for (int f = tid; f < CHUNK * (DIM / 4); f += 128) {
        int row = f >> 4, c4 = (f & 15) * 4;
        async_copy_b128((unsigned)(uintptr_t)&kc[0][row][c4],
                        kbatch + row * DIM + c4);
    }

    // ---- load query tile into LDS (coalesced float4) ----
    const float* qbase = q + ((size_t)b * NQ + q0) * DIM;
    for (int f = tid; f < QTILE * (DIM / 4); f += 128) {
        int row = f >> 4, c4 = (f & 15) * 4;
        *(v4f*)&qs[row][c4] = *(const v4f*)(qbase + row * DIM + c4);
    }
    __syncthreads();

    // ---- preload A-matrix fragments (constant for whole kernel) ----
    // WMMA f32 16x16x4 A layout: lane L holds row M=L%16, K = k0 + 2*(L/16) .. +1
    const int r    = lane & 15;
    const int koff = (lane >> 4) * 2;
    v2f afrag[16];
#pragma unroll
    for (int k = 0; k < 16; ++k)
        afrag[k] = *(const v2f*)&qs[r][k * 4 + koff];

    // per-wave selection state: 4 queries per wave, sorted ascending across lanes
    float tval[4];
    int   tidx[4];
#pragma unroll
    for (int j = 0; j < 4; ++j) { tval[j] = -__builtin_inff(); tidx[j] = 0; }

    for (int ci = 0; ci < NCHUNK; ++ci) {
        const int cur = ci & 1;
        const int mb  = ci * CHUNK;

        wait_async0();        // my async copies for chunk ci have landed in LDS
        __syncthreads();      // everyone's copies landed; prev sc readers done

        // kick off async copy of the NEXT chunk into the other buffer
        if (ci + 1 < NCHUNK) {
            const float* kb2 = kbatch + (size_t)(mb + CHUNK) * DIM;
            for (int f = tid; f < CHUNK * (DIM / 4); f += 128) {
                int row = f >> 4, c4 = (f & 15) * 4;
                async_copy_b128((unsigned)(uintptr_t)&kc[cur ^ 1][row][c4],
                                kb2 + row * DIM + c4);
            }
        }

        // ---- WMMA: wave w computes keys [16w, 16w+16) for all 16 queries ----
        // two interleaved accumulation chains to keep the matrix pipe busy
        const int nb = wave * 16;
        const int n  = nb + r;               // B layout mirrors A (row = lane%16)
        v8f acc0 = {}, acc1 = {};
#pragma unroll
        for (int k = 0; k < 16; k += 2) {
            v2f b0 = *(const v2f*)&kc[cur][n][k * 4 + koff];
            v2f b1 = *(const v2f*)&kc[cur][n][(k + 1) * 4 + koff];
            acc0 = __builtin_amdgcn_wmma_f32_16x16x4_f32(
                false, afrag[k], false, b0, (short)0, acc0, false, false);
            acc1 = __builtin_amdgcn_wmma_f32_16x16x4_f32(
                false, afrag[k + 1], false, b1, (short)0, acc1, false, false);
        }
        // C/D layout: VGPR i, lanes 0-15 -> q=i ; lanes 16-31 -> q=i+8
        const int qrow = (lane >> 4) * 8;
#pragma unroll
        for (int i = 0; i < 8; ++i)
            sc[qrow + i][nb + r] = acc0[i] + acc1[i];
        __syncthreads();

        // ---- fused top-32 selection (wave w owns queries 4w..4w+3) ----
#pragma unroll
        for (int j = 0; j < 4; ++j) {
            const int qq = wave * 4 + j;
#pragma unroll
            for (int base = 0; base < CHUNK; base += 32) {
                float cv = sc[qq][base + lane];
                int   ci2 = mb + base + lane;
                float thr = __shfl(tval[j], 0);     // lane0 = current min of top-32
                if (__ballot(cv > thr) == 0) continue;  // wave-uniform early out

                // bitonic sort candidates ascending across 32 lanes
#pragma unroll
                for (int kk = 2; kk <= 32; kk <<= 1) {
#pragma unroll
                    for (int jj = kk >> 1; jj > 0; jj >>= 1) {
                        float pv = __shfl_xor(cv, jj);
                        int   pi = __shfl_xor(ci2, jj);
                        bool up      = ((lane & kk) == 0);
                        bool lower   = ((lane & jj) == 0);
                        bool keepMin = (up == lower);
                        bool take    = keepMin ? (pv < cv) : (pv > cv);
                        float nv = take ? pv : cv;
                        int   ni = take ? pi : ci2;
                        cv = nv; ci2 = ni;
                    }
                }
                // reverse -> descending
                cv  = __shfl_xor(cv, 31);
                ci2 = __shfl_xor(ci2, 31);

                // pairwise max of (ascending current, descending candidates):
                // top-32 of the union, as a bitonic sequence
                bool cg  = cv > tval[j];
                float mv = cg ? cv : tval[j];
                int   mi = cg ? ci2 : tidx[j];

                // bitonic merge -> ascending
#pragma unroll
                for (int jj = 16; jj > 0; jj >>= 1) {
                    float pv = __shfl_xor(mv, jj);
                    int   pi = __shfl_xor(mi, jj);
                    bool lower = ((lane & jj) == 0);
                    bool take  = lower ? (pv < mv) : (pv > mv);
                    float nv = take ? pv : mv;
                    int   ni = take ? pi : mi;
                    mv = nv; mi = ni;
                }
                tval[j] = mv; tidx[j] = mi;
            }
        }
        // no trailing barrier: next iteration's top barrier orders sc reuse
    }

    // ---- publish sorted (ascending) top-32 indices ----
    __syncthreads();
#pragma unroll
    for (int j = 0; j < 4; ++j)
        top_idx[wave * 4 + j][lane] = tidx[j];
    __syncthreads();

    // ---- gather db[b, idx] : 128 contiguous floats per (q, rank) ----
    // one wave per pair: 32 lanes x float4 = 128 floats, fully coalesced
    float*       kvb = out_kv + ((size_t)b * NQ + q0) * TOPK * 2 * DIM;
    const float* dbb = db + (size_t)b * MKEYS * 2 * DIM;
    for (int p = wave; p < QTILE * TOPK; p += 4) {
        int qq = p >> 5;
        int rk = p & 31;
        int mi = top_idx[qq][rk];
        v4f v = *(const v4f*)(dbb + (size_t)mi * 2 * DIM + lane * 4);
        *(v4f*)(kvb + (size_t)p * 2 * DIM + lane * 4) = v;
    }

    // ---- mask (always valid) ----
    float* mout = out_mask + ((size_t)b * NQ + q0) * TOPK;
    for (int f = tid; f < QTILE * TOPK; f += 128)
        mout[f] = 1.0f;
}

extern "C" void kernel_launch(void* const* d_in, const int* in_sizes, int n_in,
                              void* d_out, int out_size, void* d_ws, size_t ws_size,
                              hipStream_t stream) {
    const float* q    = (const float*)d_in[0];
    const float* keys = (const float*)d_in[1];
    const float* db   = (const float*)d_in[2];
    // d_in[3] is topk (==32), fixed at compile time

    float* kv   = (float*)d_out;
    float* mask = kv + (size_t)BATCH * NQ * TOPK * 2 * DIM;

    dim3 grid(BATCH * (NQ / QTILE));   // 512 workgroups
    dim3 block(128);                   // 4 waves (wave32)
    knn_topk_kernel<<<grid, block, 0, stream>>>(q, keys, db, kv, mask);
}